// BatchTopK_111669149796
// MI455X (gfx1250) — compile-verified
//
#include <hip/hip_runtime.h>

// ---------------------------------------------------------------------------
// BatchTopK for MI455X (gfx1250): exact global top-(k*B) via 2-level radix
// select on float bit patterns. Bandwidth-bound: ~100MB in + ~100MB out HBM,
// intermediate re-reads served from the 192MB L2. K1 streams x into LDS with
// the CDNA5 Tensor Data Mover (tensor_load_to_lds + s_wait_tensorcnt).
// ---------------------------------------------------------------------------

typedef unsigned int  u32;
typedef unsigned int  u32x4 __attribute__((ext_vector_type(4)));
typedef int           i32x8 __attribute__((ext_vector_type(8)));
typedef int           i32x4 __attribute__((ext_vector_type(4)));

#define CHUNK      4096      // floats per TDM tile (16 KB)
#define K1_BLOCKS  512
#define NBINS      4096      // top-12-bit bins
#define NREP       2         // LDS histogram replicas
#define NBINS2     524288    // 2^19 low-bit bins

#if __has_builtin(__builtin_amdgcn_tensor_load_to_lds) && __has_builtin(__builtin_amdgcn_s_wait_tensorcnt)
#define USE_TDM 1
#else
#define USE_TDM 0
#endif

__device__ __forceinline__ u32 monotone_key(float v) {
  // relu + monotonic u32 key: non-positive -> 0, positive fp32 bits compare as u32
  u32 u = __float_as_uint(v);
  return (v > 0.0f) ? u : 0u;
}

#if USE_TDM
__device__ __forceinline__ u32 lds_byte_off(const void* p) {
  // generic shared pointer: low 32 bits are the LDS byte address (aperture map)
  return (u32)(unsigned long long)p;
}

__device__ __forceinline__ void tdm_load_chunk(const float* gptr, u32 lds_off, int nelem) {
  unsigned long long ga = (unsigned long long)gptr;
  u32 td0 = (u32)nelem;
  // D# group0: count=1 | lds_addr | global_addr[56:0] | type=2
  u32x4 g0;
  g0[0] = 1u;                                   // count=1, is_restore=0, gather off
  g0[1] = lds_off;                              // LDS dest byte address
  g0[2] = (u32)(ga & 0xFFFFFFFFull);            // global_addr[31:0]
  g0[3] = (u32)((ga >> 32) & 0x1FFFFFFull) | (2u << 30);  // global_addr[56:32] | type=2
  // D# group1: data_size=4B, 1-D tile of nelem elements
  i32x8 g1;
  g1[0] = (int)(2u << 16);                      // wg_mask=0, data_size=2 (4B)
  g1[1] = (int)((td0 & 0xFFFFu) << 16);         // tensor_dim0[15:0] (atomic_barrier_addr=0)
  g1[2] = (int)(((td0 >> 16) & 0xFFFFu) | (1u << 16)); // tensor_dim0[31:16] | tensor_dim1=1
  g1[3] = (int)((td0 & 0xFFFFu) << 16);         // tile_dim0 = nelem
  g1[4] = 0;                                    // tile_dim1=0 (unused), tile_dim2=0
  g1[5] = (int)td0;                             // tensor_dim0_stride lo32
  g1[6] = (int)((td0 & 0xFFFFu) << 16);         // stride0 hi16=0 | stride1 lo16
  g1[7] = 0;                                    // stride1 hi
  i32x4 gz4 = {0, 0, 0, 0};
  i32x8 gz8 = {0, 0, 0, 0, 0, 0, 0, 0};
  // clang-23 / therock-10.0 signature: 6 args (g0, g1, g2, g3, pad, cpol)
  __builtin_amdgcn_tensor_load_to_lds(g0, g1, gz4, gz4, gz8, 0);
}
#endif

// --------------------------- K0: zero hist2 --------------------------------
__global__ void bk_zero(u32* __restrict__ hist2) {
  int i = blockIdx.x * blockDim.x + threadIdx.x;   // 131072 threads, uint4 each
  u32* p = hist2 + (size_t)i * 4;
  p[0] = 0u; p[1] = 0u; p[2] = 0u; p[3] = 0u;
}

// --------------- K1: coarse 12-bit histogram (TDM-fed LDS) -----------------
__global__ __launch_bounds__(1024) void bk_hist1(const float* __restrict__ x,
                                                 u32* __restrict__ partial,
                                                 int chunksPerBlock) {
  __shared__ u32 lh[NBINS * NREP];
#if USE_TDM
  __shared__ float tile[2][CHUNK];
#endif
  const int tid = threadIdx.x;
  for (int i = tid; i < NBINS * NREP; i += 1024) lh[i] = 0u;
  __syncthreads();

  const long long base = (long long)blockIdx.x * chunksPerBlock;
#if USE_TDM
  if (tid < 32) {
    tdm_load_chunk(x + base * CHUNK, lds_byte_off(&tile[0][0]), CHUNK);
  }
#endif
  for (int c = 0; c < chunksPerBlock; ++c) {
#if USE_TDM
    if (tid < 32) {
      if (c + 1 < chunksPerBlock) {
        tdm_load_chunk(x + (base + c + 1) * CHUNK,
                       lds_byte_off(&tile[(c + 1) & 1][0]), CHUNK);
        __builtin_amdgcn_s_wait_tensorcnt(1);   // oldest tile (c) complete
      } else {
        __builtin_amdgcn_s_wait_tensorcnt(0);
      }
    }
    __syncthreads();
    const float* src = &tile[c & 1][0];
#else
    const float* src = x + (base + c) * CHUNK;
#endif
#pragma unroll
    for (int j = 0; j < CHUNK / 1024; ++j) {
      u32 key = monotone_key(src[j * 1024 + tid]);
      atomicAdd(&lh[(key >> 19) * NREP + (tid & (NREP - 1))], 1u);
    }
    __syncthreads();   // tile safe to overwrite two iterations later
  }

  for (int b = tid; b < NBINS; b += 1024) {
    u32 s = 0;
#pragma unroll
    for (int r = 0; r < NREP; ++r) s += lh[b * NREP + r];
    partial[(size_t)blockIdx.x * NBINS + b] = s;
  }
}

// ----------------- K1b: reduce per-block partials -> hist1 -----------------
__global__ void bk_reduce1(const u32* __restrict__ partial, u32* __restrict__ hist1) {
  int b = blockIdx.x * 256 + threadIdx.x;          // 16 blocks x 256 = 4096 bins
  u32 s = 0;
  for (int i = 0; i < K1_BLOCKS; ++i) s += partial[(size_t)i * NBINS + b];
  hist1[b] = s;
}

// -------------- K2: suffix-scan hist1, find coarse bin B -------------------
__global__ __launch_bounds__(1024) void bk_scan1(const u32* __restrict__ hist1,
                                                 const int* __restrict__ kptr,
                                                 int batch,
                                                 u32* __restrict__ scal) {
  __shared__ u32 ss[1024];
  const int t = threadIdx.x;
  u32 h0 = hist1[t * 4 + 0], h1 = hist1[t * 4 + 1];
  u32 h2 = hist1[t * 4 + 2], h3 = hist1[t * 4 + 3];
  ss[t] = h0 + h1 + h2 + h3;
  __syncthreads();
  for (int off = 1; off < 1024; off <<= 1) {
    u32 v = (t + off < 1024) ? ss[t + off] : 0u;
    __syncthreads();
    ss[t] += v;
    __syncthreads();
  }
  const u32 nk = (u32)(kptr[0] * batch);
  u32 mine = ss[t];
  u32 next = (t < 1023) ? ss[t + 1] : 0u;
  if (mine >= nk && next < nk) {                   // crossing lies in my 4 bins
    u32 suf[5];
    suf[0] = mine;
    suf[1] = suf[0] - h0;
    suf[2] = suf[1] - h1;
    suf[3] = suf[2] - h2;
    suf[4] = next;
#pragma unroll
    for (int j = 0; j < 4; ++j) {
      if (suf[j] >= nk && suf[j + 1] < nk) {
        scal[0] = (u32)(t * 4 + j);                // coarse bin B
        scal[1] = suf[j + 1];                      // count of keys strictly above bin B
      }
    }
  }
}

// ----------- K3: fine 19-bit histogram for elements in bin B ---------------
__global__ void bk_hist2(const float4* __restrict__ x4,
                         const u32* __restrict__ scal,
                         u32* __restrict__ hist2, int total4) {
  const u32 B = scal[0];
  const int stride = gridDim.x * blockDim.x;
  for (int i = blockIdx.x * blockDim.x + threadIdx.x; i < total4; i += stride) {
    float4 v = x4[i];
    u32 k0 = monotone_key(v.x), k1 = monotone_key(v.y);
    u32 k2 = monotone_key(v.z), k3 = monotone_key(v.w);
    if ((k0 >> 19) == B) atomicAdd(&hist2[k0 & 0x7FFFFu], 1u);
    if ((k1 >> 19) == B) atomicAdd(&hist2[k1 & 0x7FFFFu], 1u);
    if ((k2 >> 19) == B) atomicAdd(&hist2[k2 & 0x7FFFFu], 1u);
    if ((k3 >> 19) == B) atomicAdd(&hist2[k3 & 0x7FFFFu], 1u);
  }
}

// ---------- K4: suffix-scan hist2, exact threshold key + tie quota ---------
__global__ __launch_bounds__(1024) void bk_scan2(const u32* __restrict__ hist2,
                                                 const int* __restrict__ kptr,
                                                 int batch,
                                                 u32* __restrict__ scal) {
  __shared__ u32 ss[1024];
  const int t = threadIdx.x;
  const uint4* h4 = (const uint4*)(hist2 + (size_t)t * 512);
  u32 s = 0;
  for (int i = 0; i < 128; ++i) { uint4 q = h4[i]; s += q.x + q.y + q.z + q.w; }
  ss[t] = s;
  __syncthreads();
  for (int off = 1; off < 1024; off <<= 1) {
    u32 v = (t + off < 1024) ? ss[t + off] : 0u;
    __syncthreads();
    ss[t] += v;
    __syncthreads();
  }
  const u32 nk = (u32)(kptr[0] * batch);
  const u32 need = nk - scal[1];                   // survivors needed within bin B
  u32 mine = ss[t];
  u32 next = (t < 1023) ? ss[t + 1] : 0u;
  if (mine >= need && next < need) {
    u32 running = next;                            // suffix strictly above current sub-bin
    for (int i = 511; i >= 0; --i) {
      u32 c = hist2[t * 512 + i];
      if (running + c >= need) {
        u32 T19 = (u32)(t * 512 + i);
        scal[2] = (scal[0] << 19) | T19;           // exact 31-bit threshold key
        scal[4] = need - running;                  // #ties at threshold to keep (>=1)
        break;
      }
      running += c;
    }
  }
}

// ------------------- K5: write masked output (100MB) -----------------------
__global__ void bk_write(const float4* __restrict__ x4,
                         u32* __restrict__ scal,
                         float4* __restrict__ out4, int total4) {
  const u32 T = scal[2];
  int* tie = (int*)&scal[4];
  const int stride = gridDim.x * blockDim.x;
  for (int i = blockIdx.x * blockDim.x + threadIdx.x; i < total4; i += stride) {
    float4 v = x4[i];
    float4 o;
    {
      u32 k = monotone_key(v.x);
      o.x = (k > T) ? v.x : ((k == T) ? ((atomicSub(tie, 1) > 0) ? v.x : 0.0f) : 0.0f);
    }
    {
      u32 k = monotone_key(v.y);
      o.y = (k > T) ? v.y : ((k == T) ? ((atomicSub(tie, 1) > 0) ? v.y : 0.0f) : 0.0f);
    }
    {
      u32 k = monotone_key(v.z);
      o.z = (k > T) ? v.z : ((k == T) ? ((atomicSub(tie, 1) > 0) ? v.z : 0.0f) : 0.0f);
    }
    {
      u32 k = monotone_key(v.w);
      o.w = (k > T) ? v.w : ((k == T) ? ((atomicSub(tie, 1) > 0) ? v.w : 0.0f) : 0.0f);
    }
    out4[i] = o;
  }
}

// ---------------------------------------------------------------------------
extern "C" void kernel_launch(void* const* d_in, const int* in_sizes, int n_in,
                              void* d_out, int out_size, void* d_ws, size_t ws_size,
                              hipStream_t stream) {
  (void)n_in; (void)out_size; (void)ws_size;
  const float* x    = (const float*)d_in[0];
  const int*   kptr = (const int*)d_in[1];
  float*       out  = (float*)d_out;

  const int total  = in_sizes[0];                  // 25,165,824
  const int batch  = total / 24576;                // 1024 (reference BATCH)
  const int total4 = total / 4;

  u32* ws      = (u32*)d_ws;
  u32* hist2   = ws;                               // 524288 u32 (2 MB)
  u32* hist1   = ws + NBINS2;                      // 4096 u32
  u32* partial = hist1 + NBINS;                    // 512*4096 u32 (8 MB)
  u32* scal    = partial + (size_t)K1_BLOCKS * NBINS;  // scalars

  const int nchunks = total / CHUNK;               // 6144
  const int cpb     = nchunks / K1_BLOCKS;         // 12

  bk_zero  <<<NBINS2 / (4 * 256), 256, 0, stream>>>(hist2);
  bk_hist1 <<<K1_BLOCKS, 1024, 0, stream>>>(x, partial, cpb);
  bk_reduce1<<<NBINS / 256, 256, 0, stream>>>(partial, hist1);
  bk_scan1 <<<1, 1024, 0, stream>>>(hist1, kptr, batch, scal);
  bk_hist2 <<<8192, 256, 0, stream>>>((const float4*)x, scal, hist2, total4);
  bk_scan2 <<<1, 1024, 0, stream>>>(hist2, kptr, batch, scal);
  bk_write <<<8192, 256, 0, stream>>>((const float4*)x, scal, (float4*)out, total4);
}